// HardDetectionModule_12687333392421
// MI455X (gfx1250) — compile-verified
//
#include <hip/hip_runtime.h>
#include <hip/hip_bf16.h>
#include <stdint.h>

// ---------------- problem constants (match reference) ----------------
#define BB 16
#define CC 512
#define HH 80
#define WW 80
#define HWPIX (HH * WW)            // 6400 pixels per plane
#define NPLANES (BB * CC)          // 8192 (b,c) planes
#define WORDS_PER_PLANE (HWPIX / 32)  // 200 bitmask words per plane
#define N_MAX (BB * HWPIX)         // 102400 padded detections
#define OUT_TOTAL (4 * N_MAX)      // grid_keypoints(2*N) + keypoints(2*N)

// ---------------- CDNA5 async global->LDS path (guarded) ----------------
// gfx1250 GLOBAL_LOAD_ASYNC_TO_LDS_B128, tracked by ASYNCcnt. The clang-22
// builtin signature (leaked by the round-1 diagnostic) is
//   (v4i addrspace(1)* src, v4i addrspace(3)* dst, imm int offset, imm int cpol)
// Low 32 bits of a generic shared pointer are the LDS byte offset (ISA 10.2);
// for global memory the generic VA equals the AS1 VA.
#if __has_builtin(__builtin_amdgcn_global_load_async_to_lds_b128)
#define CDNA5_ASYNC_LDS 1
#else
#define CDNA5_ASYNC_LDS 0
#endif

#if CDNA5_ASYNC_LDS
typedef int cdna5_v4i __attribute__((ext_vector_type(4)));
typedef __attribute__((address_space(1))) cdna5_v4i* cdna5_g4p;
typedef __attribute__((address_space(3))) cdna5_v4i* cdna5_l4p;
#endif

__device__ __forceinline__ void async_copy_b128(const void* gsrc, void* ldst) {
#if CDNA5_ASYNC_LDS
    __builtin_amdgcn_global_load_async_to_lds_b128(
        (cdna5_g4p)(unsigned long long)(uintptr_t)gsrc,
        (cdna5_l4p)(unsigned int)(uintptr_t)ldst,
        0, 0);
#else
    *(float4*)ldst = *(const float4*)gsrc;
#endif
}

__device__ __forceinline__ void async_copy_wait() {
#if CDNA5_ASYNC_LDS
#if __has_builtin(__builtin_amdgcn_s_wait_asynccnt)
    __builtin_amdgcn_s_wait_asynccnt(0);
#else
    asm volatile("s_wait_asynccnt 0" ::: "memory");
#endif
#endif
}

// ---------------- stage 1: init cmax bits + pad outputs ----------------
__global__ __launch_bounds__(256) void init_kernel(unsigned* __restrict__ cmaxU,
                                                   float* __restrict__ out) {
    int i = blockIdx.x * 256 + threadIdx.x;
    if (i < N_MAX) cmaxU[i] = 0u;                 // relu >= 0 -> 0 is identity for max
    if (i < 2 * N_MAX) out[i] = -1.0f;            // grid_keypoints fill
    else if (i < OUT_TOTAL) out[i] = -4.5f;       // keypoints fill = 8*(-1)+3.5
}

// ---------------- stage 2: channel-wise max of relu(x) ----------------
// grid = (6400/256 pixel tiles, 512/64 channel chunks, 16 batches)
__global__ __launch_bounds__(256) void cmax_kernel(const float* __restrict__ x,
                                                   unsigned* __restrict__ cmaxU) {
    const int pix = blockIdx.x * 256 + threadIdx.x;   // 0..6399
    const int c0  = blockIdx.y * 64;
    const int b   = blockIdx.z;
    const float* base = x + ((size_t)b * CC + c0) * HWPIX + pix;
    float m = 0.0f;                                   // relu lower bound
#pragma unroll 8
    for (int c = 0; c < 64; ++c)
        m = fmaxf(m, base[(size_t)c * HWPIX]);
    // non-negative IEEE floats: uint compare == float compare
    atomicMax(&cmaxU[b * HWPIX + pix], __float_as_uint(m));
}

// ---------------- stage 3: per-plane detection bitmask + count ----------------
__global__ __launch_bounds__(256) void detect_kernel(const float* __restrict__ x,
                                                     const unsigned* __restrict__ cmaxU,
                                                     unsigned* __restrict__ bitmask,
                                                     unsigned* __restrict__ counts) {
    __shared__ __align__(16) float tile[HWPIX];   // relu(x) plane
    __shared__ __align__(16) float ctile[HWPIX];  // channel max per pixel
    __shared__ unsigned red[256];

    const int plane = blockIdx.x;        // = b*512 + c
    const int b = plane >> 9;
    const int t = threadIdx.x;

    const float* gp = x + (size_t)plane * HWPIX;
    const float* gc = (const float*)cmaxU + (size_t)b * HWPIX;

    for (int i = t; i < HWPIX / 4; i += 256) {
        async_copy_b128(gp + i * 4, &tile[i * 4]);
        async_copy_b128(gc + i * 4, &ctile[i * 4]);
    }
    async_copy_wait();
    __syncthreads();

    for (int i = t; i < HWPIX; i += 256) tile[i] = fmaxf(tile[i], 0.0f);
    __syncthreads();

    unsigned word = 0u, cnt = 0u;
    if (t < WORDS_PER_PLANE) {
        const int p0 = t * 32;
        for (int j = 0; j < 32; ++j) {
            const int p = p0 + j;
            const int h = p / WW, w = p % WW;
            const float v = tile[p];
            bool det = (v == ctile[p]);           // depth-wise max (exact ==)
            if (det) {
                const int h0 = (h > 0) ? h - 1 : h, h1 = (h < HH - 1) ? h + 1 : h;
                const int w0 = (w > 0) ? w - 1 : w, w1 = (w < WW - 1) ? w + 1 : w;
                for (int hh = h0; hh <= h1; ++hh)
                    for (int ww = w0; ww <= w1; ++ww)
                        if (tile[hh * WW + ww] > v) det = false;   // 3x3 local max
            }
            word |= (det ? 1u : 0u) << j;
        }
        cnt = __popc(word);
        bitmask[(size_t)plane * WORDS_PER_PLANE + t] = word;
    }
    red[t] = cnt;
    __syncthreads();
    for (int s = 128; s > 0; s >>= 1) {
        if (t < s) red[t] += red[t + s];
        __syncthreads();
    }
    if (t == 0) counts[plane] = red[0];
}

// ---------------- stage 4: exclusive scan of 8192 plane counts ----------------
__global__ __launch_bounds__(256) void scan_kernel(const unsigned* __restrict__ counts,
                                                   unsigned* __restrict__ offsets) {
    __shared__ unsigned s[256];
    const int t = threadIdx.x;
    const int base = t * 32;
    unsigned loc[32];
    unsigned sum = 0u;
#pragma unroll
    for (int i = 0; i < 32; ++i) { loc[i] = counts[base + i]; sum += loc[i]; }
    s[t] = sum;
    __syncthreads();
    for (int off = 1; off < 256; off <<= 1) {
        unsigned v = (t >= off) ? s[t - off] : 0u;
        __syncthreads();
        s[t] += v;
        __syncthreads();
    }
    unsigned run = s[t] - sum;   // exclusive base for this thread's segment
#pragma unroll
    for (int i = 0; i < 32; ++i) { offsets[base + i] = run; run += loc[i]; }
}

// ---------------- stage 5: ordered compaction into outputs ----------------
__global__ __launch_bounds__(256) void compact_kernel(const unsigned* __restrict__ bitmask,
                                                      const unsigned* __restrict__ offsets,
                                                      float* __restrict__ out) {
    __shared__ unsigned s[256];
    const int plane = blockIdx.x;
    const int t = threadIdx.x;

    unsigned word = 0u, cnt = 0u;
    if (t < WORDS_PER_PLANE) {
        word = bitmask[(size_t)plane * WORDS_PER_PLANE + t];
        cnt = __popc(word);
    }
    s[t] = cnt;
    __syncthreads();
    for (int off = 1; off < 256; off <<= 1) {
        unsigned v = (t >= off) ? s[t - off] : 0u;
        __syncthreads();
        s[t] += v;
        __syncthreads();
    }
    if (t < WORDS_PER_PLANE && word) {
        unsigned n = offsets[plane] + s[t] - cnt;   // global ordered slot
        const int p0 = t * 32;
        unsigned wrd = word;
        while (wrd) {
            const int j = __ffs(wrd) - 1;           // LSB first -> ascending pixel
            wrd &= wrd - 1u;
            if (n < (unsigned)N_MAX) {
                const int p = p0 + j;
                const float fh = (float)(p / WW);
                const float fw = (float)(p % WW);
                out[2u * n]             = fh;
                out[2u * n + 1u]        = fw;
                out[2u * (unsigned)N_MAX + 2u * n]      = fh * 8.0f + 3.5f;
                out[2u * (unsigned)N_MAX + 2u * n + 1u] = fw * 8.0f + 3.5f;
            }
            ++n;
        }
    }
}

// ---------------- launch ----------------
extern "C" void kernel_launch(void* const* d_in, const int* in_sizes, int n_in,
                              void* d_out, int out_size, void* d_ws, size_t ws_size,
                              hipStream_t stream) {
    const float* x = (const float*)d_in[0];
    float* out = (float*)d_out;

    char* ws = (char*)d_ws;
    unsigned* cmaxU   = (unsigned*)ws;                                   // 102400 u32
    unsigned* counts  = (unsigned*)(ws + 409600);                        // 8192 u32
    unsigned* offsets = (unsigned*)(ws + 409600 + 32768);                // 8192 u32
    unsigned* bitmask = (unsigned*)(ws + 409600 + 65536);                // 8192*200 u32

    init_kernel<<<(OUT_TOTAL + 255) / 256, 256, 0, stream>>>(cmaxU, out);
    cmax_kernel<<<dim3(HWPIX / 256, CC / 64, BB), 256, 0, stream>>>(x, cmaxU);
    detect_kernel<<<NPLANES, 256, 0, stream>>>(x, cmaxU, bitmask, counts);
    scan_kernel<<<1, 256, 0, stream>>>(counts, offsets);
    compact_kernel<<<NPLANES, 256, 0, stream>>>(bitmask, offsets, out);
}